// MaskTypeProbabilitiesLayer_2147483648370
// MI455X (gfx1250) — compile-verified
//
#include <hip/hip_runtime.h>
#include <stdint.h>

// Problem constants (from reference): B=2048, L=8191, F=8, out = [B, L, 8] f32.
#define SEQ_L   8191
#define FEAT    8
#define THREADS 1024
#define PER_THR 8   // ceil(8191/1024)

typedef __attribute__((ext_vector_type(4))) unsigned int u32x4;
typedef __attribute__((ext_vector_type(8))) int          i32x8;
typedef __attribute__((ext_vector_type(4))) int          i32x4;

__global__ __launch_bounds__(THREADS)
void mask_type_prob_kernel(const float* __restrict__ song, float* __restrict__ out)
{
    __shared__ float lds_types[SEQ_L + 1];   // 32768 B staging for the type column
    __shared__ int   s_first5, s_first6;

    const int b = blockIdx.x;
    const float* row = song + (size_t)b * ((size_t)SEQ_L * FEAT);

    // ---- Stage the strided type column into LDS with the Tensor Data Mover ----
    if (threadIdx.x == 0) {
        s_first5 = 0x7FFFFFFF;
        s_first6 = 0x7FFFFFFF;

        const uint64_t gaddr = (uint64_t)(uintptr_t)row;
        const uint32_t laddr = (uint32_t)(uintptr_t)(&lds_types[0]); // LDS byte offset (addr[31:0])

        // D# group 0 (128b): count=1 | lds_addr | global_addr[56:0] | type=2
        u32x4 g0;
        g0[0] = 1u;                                                   // count=1, user mode
        g0[1] = laddr;                                                // lds_addr
        g0[2] = (uint32_t)gaddr;                                      // global_addr[31:0]
        g0[3] = (uint32_t)((gaddr >> 32) & 0x01FFFFFFull) | (2u << 30); // addr[56:32] | type=2

        // D# group 1 (256b): 2D tensor (dim0=FEAT elems, stride FEAT), tile = 1 x SEQ_L
        i32x8 g1;
        g1[0] = (int)(2u << 16);        // workgroup_mask=0, data_size=2 (4 bytes)
        g1[1] = (int)(FEAT  << 16);     // atomic_barrier_addr=0 | tensor_dim0[15:0]=8
        g1[2] = (int)(SEQ_L << 16);     // tensor_dim0[31:16]=0  | tensor_dim1[15:0]=8191
        g1[3] = (int)(1u    << 16);     // tensor_dim1[31:16]=0  | tile_dim0=1
        g1[4] = (int)SEQ_L;             // tile_dim1=8191        | tile_dim2=0
        g1[5] = (int)FEAT;              // tensor_dim0_stride[31:0]=8
        g1[6] = 0;                      // dim0_stride[47:32]=0  | dim1_stride[15:0]=0
        g1[7] = 0;                      // dim1_stride[47:16]=0

        i32x4 gz = {0, 0, 0, 0};        // groups 2/3 unused (tile_dim2/3 = 0)
#if defined(__clang_major__) && (__clang_major__ >= 23)
        i32x8 gz8 = {0, 0, 0, 0, 0, 0, 0, 0};
        __builtin_amdgcn_tensor_load_to_lds(g0, g1, gz, gz, gz8, 0);
#else
        __builtin_amdgcn_tensor_load_to_lds(g0, g1, gz, gz, 0);
#endif
        __builtin_amdgcn_s_wait_tensorcnt(0);
    }
    __syncthreads();

    // ---- Per-thread: read 8 types from LDS, track first index of 5 / 6 ----
    int loc5 = 0x7FFFFFFF, loc6 = 0x7FFFFFFF;
    int ty[PER_THR];
#pragma unroll
    for (int k = 0; k < PER_THR; ++k) {
        const int i = (int)threadIdx.x + k * THREADS;
        const float fv = (i < SEQ_L) ? lds_types[i] : -1.0f;
        const int t = (int)fv;
        ty[k] = t;
        if (t == 5) loc5 = min(loc5, i);
        if (t == 6) loc6 = min(loc6, i);
    }

    // wave32 min-reduction, then one LDS atomic per wave
#pragma unroll
    for (int off = 16; off > 0; off >>= 1) {
        loc5 = min(loc5, __shfl_xor(loc5, off, 32));
        loc6 = min(loc6, __shfl_xor(loc6, off, 32));
    }
    if ((threadIdx.x & 31u) == 0u) {
        atomicMin(&s_first5, loc5);
        atomicMin(&s_first6, loc6);
    }
    __syncthreads();

    const int first5 = s_first5;
    const int first6 = s_first6;

    // ---- Emit 8 floats per position, two coalesced float4 stores ----
    float* orow = out + (size_t)b * ((size_t)SEQ_L * 8);
#pragma unroll
    for (int k = 0; k < PER_THR; ++k) {
        const int i = (int)threadIdx.x + k * THREADS;
        if (i < SEQ_L) {
            const int t = ty[k];
            unsigned m;
            if (t >= 4 && t <= 6) {
                m = (i < first5) ? 0x20u : ((i < first6) ? 0x40u : 0xF8u);
            } else {
                // type 0:0x02  1:0x06  2:0x10  3:0xF8  7:0x80  8:0x00
                m = (t < 4) ? ((0xF8100602u >> (t * 8)) & 0xFFu)
                            : ((t == 7) ? 0x80u : 0x00u);
            }
            float4 v0, v1;
            v0.x = (m & 0x01u) ? 1.f : 0.f;
            v0.y = (m & 0x02u) ? 1.f : 0.f;
            v0.z = (m & 0x04u) ? 1.f : 0.f;
            v0.w = (m & 0x08u) ? 1.f : 0.f;
            v1.x = (m & 0x10u) ? 1.f : 0.f;
            v1.y = (m & 0x20u) ? 1.f : 0.f;
            v1.z = (m & 0x40u) ? 1.f : 0.f;
            v1.w = (m & 0x80u) ? 1.f : 0.f;
            float4* p = (float4*)(orow + (size_t)i * 8);
            p[0] = v0;
            p[1] = v1;
        }
    }
}

extern "C" void kernel_launch(void* const* d_in, const int* in_sizes, int n_in,
                              void* d_out, int out_size, void* d_ws, size_t ws_size,
                              hipStream_t stream)
{
    (void)n_in; (void)out_size; (void)d_ws; (void)ws_size;
    const float* song = (const float*)d_in[0];
    float* out = (float*)d_out;
    const int B = in_sizes[0] / (SEQ_L * FEAT);   // 2048
    hipLaunchKernelGGL(mask_type_prob_kernel, dim3(B), dim3(THREADS), 0, stream, song, out);
}